// Protein_self_attention_40235253629310
// MI455X (gfx1250) — compile-verified
//
#include <hip/hip_runtime.h>

// Protein self-attention (per-channel softmax), MI455X / gfx1250.
//
// Strategy:
//  - Q/K/V/O projections: V_WMMA_F32_16X16X4_F32 (fp32-exact, wave32,
//    one wave per 16x16 output tile, K-loop of 64 WMMAs).
//  - Attention: softmax over keys is per-(i,c) scalar with logits a*k_j,
//    a = q[i,c]/16. max_j(a*k_j) = a * (a>=0 ? kmax[c] : kmin[c]) is analytic,
//    so a single streaming pass with one v_exp_f32 + 2 FMAs per key suffices.
//    ~768*768*256 = 151M exp2 ops dominate; all data is L2-resident (~3MB).

#define DIM   256
#define NROW  768
#define LOG2E 1.4426950408889634f

typedef __attribute__((ext_vector_type(2))) float v2f;
typedef __attribute__((ext_vector_type(8))) float v8f;

// Out[i][j] = sum_k X[i][k] * W[j][k] + bias[j]
// X: NROW x DIM row-major, W: DIM x DIM row-major (nn.Linear weight).
// One wave32 computes one 16x16 tile of Out via 64x V_WMMA_F32_16X16X4_F32.
__global__ void gemm_xwt_wmma(const float* __restrict__ X,
                              const float* __restrict__ W,
                              const float* __restrict__ bias,
                              float* __restrict__ Out) {
  const int wid  = (blockIdx.x * blockDim.x + threadIdx.x) >> 5; // wave id (uniform)
  const int lane = threadIdx.x & 31;
  const int ti   = wid / (DIM / 16);       // row tile (M)
  const int tj   = wid % (DIM / 16);       // col tile (N)
  const int i0   = ti * 16, j0 = tj * 16;
  const int g    = lane >> 4;              // lane half (0/1)
  const int l    = lane & 15;

  const float* __restrict__ xrow = X + (i0 + l) * DIM; // A: M = l
  const float* __restrict__ wrow = W + (j0 + l) * DIM; // B: N = l (B[k][n] = W[n][k])

  v8f acc = {0.f, 0.f, 0.f, 0.f, 0.f, 0.f, 0.f, 0.f};

  #pragma unroll 8
  for (int kb = 0; kb < DIM; kb += 4) {
    // A 16x4 f32 layout: lane(g,l) holds K = kb+2g, kb+2g+1 for row M=l.
    // B 4x16 f32 layout: lane(g,l) holds K = kb+2g, kb+2g+1 for col N=l.
    v2f a = *(const v2f*)(xrow + kb + 2 * g);
    v2f b = *(const v2f*)(wrow + kb + 2 * g);
    acc = __builtin_amdgcn_wmma_f32_16x16x4_f32(
        /*neg_a=*/false, a, /*neg_b=*/false, b,
        /*c_mod=*/(short)0, acc, /*reuse_a=*/false, /*reuse_b=*/false);
  }

  const float bj = bias[j0 + l];
  // C/D layout: VGPR r -> (M = g*8 + r, N = l)
  #pragma unroll
  for (int r = 0; r < 8; ++r)
    Out[(i0 + g * 8 + r) * DIM + (j0 + l)] = acc[r] + bj;
}

// Per-channel max/min of K over all 768 keys (for the analytic softmax max).
__global__ void kminmax_kernel(const float* __restrict__ K,
                               float* __restrict__ kmax,
                               float* __restrict__ kmin) {
  const int c = threadIdx.x;
  float mx = -3.402823466e+38f, mn = 3.402823466e+38f;
  for (int j = 0; j < NROW; ++j) {
    float v = K[j * DIM + c];           // coalesced row reads
    mx = fmaxf(mx, v);
    mn = fminf(mn, v);
  }
  kmax[c] = mx;
  kmin[c] = mn;
}

// Single-pass per-channel softmax-attention. Block = 256 threads (one per
// channel c), BI=2 queries per block -> 384 blocks. Inner loop: one exp2 +
// 2 FMAs per (query, key); K/V rows read fully coalesced (L2-resident).
#define BI 2
__global__ void attn_kernel(const float* __restrict__ Q,
                            const float* __restrict__ K,
                            const float* __restrict__ V,
                            const float* __restrict__ kmax,
                            const float* __restrict__ kmin,
                            float* __restrict__ Mid) {
  const int c  = threadIdx.x;
  const int i0 = blockIdx.x * BI;
  const float kx = kmax[c], kn = kmin[c];

  float a2[BI], m2[BI], num[BI], den[BI];
  #pragma unroll
  for (int ii = 0; ii < BI; ++ii) {
    float a = Q[(i0 + ii) * DIM + c] * (1.0f / 16.0f);   // sqrt(256) = 16
    float m = a * (a >= 0.f ? kx : kn);                  // analytic max_j
    a2[ii] = a * LOG2E;
    m2[ii] = m * LOG2E;
    num[ii] = 0.f;
    den[ii] = 0.f;
  }

  for (int j = 0; j < NROW; ++j) {
    const float kv = K[j * DIM + c];
    const float vv = V[j * DIM + c];
    #pragma unroll
    for (int ii = 0; ii < BI; ++ii) {
      float e = __builtin_amdgcn_exp2f(fmaf(a2[ii], kv, -m2[ii])); // v_exp_f32
      den[ii] += e;
      num[ii] = fmaf(e, vv, num[ii]);
    }
  }

  #pragma unroll
  for (int ii = 0; ii < BI; ++ii)
    Mid[(i0 + ii) * DIM + c] = num[ii] / den[ii];
}

extern "C" void kernel_launch(void* const* d_in, const int* in_sizes, int n_in,
                              void* d_out, int out_size, void* d_ws, size_t ws_size,
                              hipStream_t stream) {
  const float* X  = (const float*)d_in[0];
  const float* Wq = (const float*)d_in[1];
  const float* bq = (const float*)d_in[2];
  const float* Wk = (const float*)d_in[3];
  const float* bk = (const float*)d_in[4];
  const float* Wv = (const float*)d_in[5];
  const float* bv = (const float*)d_in[6];
  const float* Wo = (const float*)d_in[7];
  const float* bo = (const float*)d_in[8];
  float* out = (float*)d_out;

  // Workspace layout (floats): Q | K | V | Mid | kmax(256) | kmin(256)
  float* ws  = (float*)d_ws;
  float* Qb  = ws;
  float* Kb  = ws + 1 * NROW * DIM;
  float* Vb  = ws + 2 * NROW * DIM;
  float* Mid = ws + 3 * NROW * DIM;
  float* kmx = ws + 4 * NROW * DIM;
  float* kmn = kmx + DIM;

  const int tiles = (NROW / 16) * (DIM / 16);  // 768 tiles
  const dim3 gblk(256);                        // 8 waves -> 8 tiles per block
  const dim3 ggrd(tiles / 8);                  // 96 blocks

  gemm_xwt_wmma<<<ggrd, gblk, 0, stream>>>(X, Wq, bq, Qb);
  gemm_xwt_wmma<<<ggrd, gblk, 0, stream>>>(X, Wk, bk, Kb);
  gemm_xwt_wmma<<<ggrd, gblk, 0, stream>>>(X, Wv, bv, Vb);
  kminmax_kernel<<<1, DIM, 0, stream>>>(Kb, kmx, kmn);
  attn_kernel<<<NROW / BI, DIM, 0, stream>>>(Qb, Kb, Vb, kmx, kmn, Mid);
  gemm_xwt_wmma<<<ggrd, gblk, 0, stream>>>(Mid, Wo, bo, out);
}